// ContactSignedDistanceHead_20813411516625
// MI455X (gfx1250) — compile-verified
//
#include <hip/hip_runtime.h>
#include <math.h>

typedef __attribute__((ext_vector_type(2))) float v2f;
typedef __attribute__((ext_vector_type(8))) float v8f;

#define B_  16
#define K_  256
#define D_  128
#define H_  64

#if defined(__has_builtin)
#if __has_builtin(__builtin_amdgcn_global_load_async_to_lds_b128)
#define HAVE_ASYNC_LDS 1
#endif
#if __has_builtin(__builtin_amdgcn_s_wait_asynccnt)
#define HAVE_WAIT_ASYNC 1
#endif
#endif

// Types matching the builtin's diagnosed signature:
//   param0: int __attribute__((vector_size(16))) __device__(AS1) *   (global src)
//   param1: LDS destination (AS3), same pointee
typedef int v4i_vs __attribute__((vector_size(4 * sizeof(int))));
typedef __attribute__((address_space(1))) v4i_vs* gv4i_p;
typedef __attribute__((address_space(3))) v4i_vs* lv4i_p;

// ---------------------------------------------------------------------------
// Kernel 1: hi = P @ W1[:D], hj = P @ W1[D:]  via v_wmma_f32_16x16x4_f32.
// One wave (32 lanes) computes a full 16-row x 64-col tile of hi or hj.
// grid = (K/16, B, 2), block = 32 (1 wave).
// ---------------------------------------------------------------------------
__global__ __launch_bounds__(32) void feat_gemm_wmma(
    const float* __restrict__ P,    // [B,K,D]
    const float* __restrict__ W1,   // [2D,H]
    float* __restrict__ hi,         // [B,K,H]
    float* __restrict__ hj)         // [B,K,H]
{
    const int it    = blockIdx.x;        // i-tile (16 rows)
    const int b     = blockIdx.y;        // batch
    const int which = blockIdx.z;        // 0 -> hi (W1 rows [0,D)), 1 -> hj (rows [D,2D))
    const int lane  = threadIdx.x;       // 0..31
    const int mrow  = lane & 15;         // A: lane -> M row ; B/C: lane -> N col
    const int khalf = lane >> 4;         // 0: K=0..1 / M=0..7 ; 1: K=2..3 / M=8..15
    const int i0    = it * 16;

    const float* Wp   = W1 + (size_t)which * D_ * H_;
    const float* Prow = P + ((size_t)(b * K_ + i0 + mrow)) * D_;
    float* Out        = which ? hj : hi;

    v8f acc0 = {0.f,0.f,0.f,0.f,0.f,0.f,0.f,0.f};
    v8f acc1 = acc0, acc2 = acc0, acc3 = acc0;

    for (int kk = 0; kk < D_; kk += 4) {
        const int ka = kk + 2 * khalf;
        // A 16x4 f32: lane(0..15)=M, VGPR0/1 = K = 2*khalf + {0,1}
        v2f a;
        a.x = Prow[ka];
        a.y = Prow[ka + 1];

        const float* Wr0 = Wp + (size_t)ka * H_;   // W1 row ka   (K = 2*khalf)
        const float* Wr1 = Wr0 + H_;               // W1 row ka+1 (K = 2*khalf+1)

        // B 4x16 f32 (layout symmetric to A): lane(0..15)=N, VGPR0/1 = K = 2*khalf+{0,1}
        v2f bv;
        bv.x = Wr0[ 0 + mrow]; bv.y = Wr1[ 0 + mrow];
        acc0 = __builtin_amdgcn_wmma_f32_16x16x4_f32(false, a, false, bv, (short)0, acc0, false, false);
        bv.x = Wr0[16 + mrow]; bv.y = Wr1[16 + mrow];
        acc1 = __builtin_amdgcn_wmma_f32_16x16x4_f32(false, a, false, bv, (short)0, acc1, false, false);
        bv.x = Wr0[32 + mrow]; bv.y = Wr1[32 + mrow];
        acc2 = __builtin_amdgcn_wmma_f32_16x16x4_f32(false, a, false, bv, (short)0, acc2, false, false);
        bv.x = Wr0[48 + mrow]; bv.y = Wr1[48 + mrow];
        acc3 = __builtin_amdgcn_wmma_f32_16x16x4_f32(false, a, false, bv, (short)0, acc3, false, false);
    }

    // C/D 16x16 f32: lane -> N ; VGPR v -> M = v + 8*(lane>>4)
    #pragma unroll
    for (int v = 0; v < 8; ++v) {
        const int m = v + 8 * khalf;
        float* orow = Out + ((size_t)(b * K_ + i0 + m)) * H_;
        orow[ 0 + mrow] = acc0[v];
        orow[16 + mrow] = acc1[v];
        orow[32 + mrow] = acc2[v];
        orow[48 + mrow] = acc3[v];
    }
}

// ---------------------------------------------------------------------------
// Kernel 2: the heavy pass. One wave per 16x16 (i,j) tile.
// hi/hj tiles staged into LDS via the gfx1250 async copy engine
// (global_load_async_to_lds_b128 + s_wait_asynccnt), falling back to
// b128 load / ds_store if unavailable. Stride-68 padding keeps the 16 j-rows
// on distinct banks of the 64-bank LDS.
// Each lane: 8 outputs, 64-deep exact-GELU dot with W2; wave shuffle-reduce
// the squared-error partial (deterministic, no float atomics).
// grid = (K/16, K/16, B), block = 32.
// ---------------------------------------------------------------------------
__device__ __forceinline__ float gelu_exact(float x) {
    return 0.5f * x * (1.0f + erff(x * 0.70710678118654752440f));
}

__global__ __launch_bounds__(32) void pair_head(
    const float* __restrict__ hi, const float* __restrict__ hj,
    const float* __restrict__ b1, const float* __restrict__ W2,
    const float* __restrict__ b2, const float* __restrict__ pos,
    float* __restrict__ sd, float* __restrict__ partial)
{
    const int jt = blockIdx.x, it = blockIdx.y, b = blockIdx.z;
    const int lane = threadIdx.x;
    const int i0 = it * 16, j0 = jt * 16;

    __shared__ float s_hi[16 * 68];   // padded: row stride 68 floats (17 float4)
    __shared__ float s_hj[16 * 68];
    __shared__ float s_b1[64];
    __shared__ float s_w2[64];

#if defined(HAVE_ASYNC_LDS)
    {   // async cache->LDS DMA: 8 x b128 per lane per tile, no VGPR round-trip
        const float4* ghi = (const float4*)(hi + ((size_t)(b * K_ + i0)) * H_);
        const float4* ghj = (const float4*)(hj + ((size_t)(b * K_ + j0)) * H_);
        float4* shi4 = (float4*)s_hi;
        float4* shj4 = (float4*)s_hj;
        for (int t = lane; t < 256; t += 32) {
            const int d = (t >> 4) * 17 + (t & 15);
            __builtin_amdgcn_global_load_async_to_lds_b128(
                (gv4i_p)(ghi + t), (lv4i_p)(shi4 + d), 0, 0);
            __builtin_amdgcn_global_load_async_to_lds_b128(
                (gv4i_p)(ghj + t), (lv4i_p)(shj4 + d), 0, 0);
        }
        if (lane < 16)
            __builtin_amdgcn_global_load_async_to_lds_b128(
                (gv4i_p)(((const float4*)b1) + lane), (lv4i_p)(((float4*)s_b1) + lane), 0, 0);
        else
            __builtin_amdgcn_global_load_async_to_lds_b128(
                (gv4i_p)(((const float4*)W2) + lane - 16), (lv4i_p)(((float4*)s_w2) + lane - 16), 0, 0);
    }
#if defined(HAVE_WAIT_ASYNC)
    __builtin_amdgcn_s_wait_asynccnt(0);
#else
    asm volatile("s_wait_asynccnt 0x0" ::: "memory");
#endif
#else
    {   // fallback: b128 loads + ds_store
        const float4* g = (const float4*)(hi + ((size_t)(b * K_ + i0)) * H_);
        float4* s4 = (float4*)s_hi;
        for (int t = lane; t < 256; t += 32) s4[(t >> 4) * 17 + (t & 15)] = g[t];
    }
    {
        const float4* g = (const float4*)(hj + ((size_t)(b * K_ + j0)) * H_);
        float4* s4 = (float4*)s_hj;
        for (int t = lane; t < 256; t += 32) s4[(t >> 4) * 17 + (t & 15)] = g[t];
    }
    if (lane < 16)       ((float4*)s_b1)[lane]      = ((const float4*)b1)[lane];
    else                 ((float4*)s_w2)[lane - 16] = ((const float4*)W2)[lane - 16];
#endif
    __syncthreads();

    const float bias2 = b2[0];
    const int n     = lane & 15;      // local j
    const int khalf = lane >> 4;
    const int j     = j0 + n;
    const float pjx = pos[((size_t)(b * K_ + j)) * 2 + 0];
    const float pjy = pos[((size_t)(b * K_ + j)) * 2 + 1];
    const float* hjrow = &s_hj[n * 68];

    float psum = 0.0f;
    #pragma unroll
    for (int v = 0; v < 8; ++v) {
        const int m = v + 8 * khalf;
        const int i = i0 + m;
        const float* hirow = &s_hi[m * 68];
        float acc = bias2;
        #pragma unroll 16
        for (int h = 0; h < 64; ++h) {
            const float t = hirow[h] + hjrow[h] + s_b1[h];
            acc = fmaf(gelu_exact(t), s_w2[h], acc);
        }
        sd[((size_t)(b * K_ + i)) * K_ + j] = acc;

        const float dx = pos[((size_t)(b * K_ + i)) * 2 + 0] - pjx;
        const float dy = pos[((size_t)(b * K_ + i)) * 2 + 1] - pjy;
        const float d2 = dx * dx + dy * dy;
        const float td = (d2 > 0.0f) ? sqrtf(d2) : 0.0f;
        const float e  = acc - td;
        psum = fmaf(e, e, psum);
    }

    // wave32 reduction
    #pragma unroll
    for (int off = 16; off > 0; off >>= 1)
        psum += __shfl_xor(psum, off, 32);
    if (lane == 0)
        partial[((size_t)blockIdx.z * gridDim.y + blockIdx.y) * gridDim.x + blockIdx.x] = psum;
}

// ---------------------------------------------------------------------------
// Kernel 3: sum 4096 partials -> mean -> loss scalar (ACH_SCALE = 1.0)
// ---------------------------------------------------------------------------
__global__ __launch_bounds__(256) void reduce_loss(
    const float* __restrict__ partial, float* __restrict__ loss)
{
    __shared__ float s[256];
    float v = 0.0f;
    for (int idx = threadIdx.x; idx < 4096; idx += 256) v += partial[idx];
    s[threadIdx.x] = v;
    __syncthreads();
    for (int off = 128; off > 0; off >>= 1) {
        if (threadIdx.x < off) s[threadIdx.x] += s[threadIdx.x + off];
        __syncthreads();
    }
    if (threadIdx.x == 0)
        loss[0] = s[0] * (1.0f / ((float)B_ * (float)K_ * (float)K_));
}

// ---------------------------------------------------------------------------
extern "C" void kernel_launch(void* const* d_in, const int* in_sizes, int n_in,
                              void* d_out, int out_size, void* d_ws, size_t ws_size,
                              hipStream_t stream) {
    (void)in_sizes; (void)n_in; (void)out_size; (void)ws_size;

    const float* P   = (const float*)d_in[0];  // particles [B,K,D]
    const float* pos = (const float*)d_in[1];  // positions [B,K,2]
    const float* W1  = (const float*)d_in[2];  // [2D,H]
    const float* b1  = (const float*)d_in[3];  // [H]
    const float* W2  = (const float*)d_in[4];  // [H,1]
    const float* b2  = (const float*)d_in[5];  // [1]

    float* sd   = (float*)d_out;                         // [B,K,K]
    float* loss = sd + (size_t)B_ * K_ * K_;             // scalar

    float* ws      = (float*)d_ws;
    float* hi      = ws;                                  // [B,K,H] = 1 MB
    float* hj      = hi + (size_t)B_ * K_ * H_;           // [B,K,H] = 1 MB
    float* partial = hj + (size_t)B_ * K_ * H_;           // [4096]

    feat_gemm_wmma<<<dim3(K_ / 16, B_, 2), 32, 0, stream>>>(P, W1, hi, hj);
    pair_head<<<dim3(K_ / 16, K_ / 16, B_), 32, 0, stream>>>(hi, hj, b1, W2, b2, pos, sd, partial);
    reduce_loss<<<1, 256, 0, stream>>>(partial, loss);
}